// Classifier_72258529788341
// MI455X (gfx1250) — compile-verified
//
#include <hip/hip_runtime.h>
#include <hip/hip_bf16.h>

#define SEQ     2048
#define DD      512
#define NB      16
#define NCHUNK  32
#define CROWS   (SEQ / NCHUNK)   // 64 rows per chunk

typedef __attribute__((ext_vector_type(2))) float v2f;
typedef __attribute__((ext_vector_type(8))) float v8f;

// ---------------------------------------------------------------------------
// Kernel 1: lengths.  len[t][b] = SEQ - sum(mask[t][b, :])   (32 tiny blocks)
// ---------------------------------------------------------------------------
__global__ void len_kernel(const int* __restrict__ mask_x,
                           const int* __restrict__ mask_y,
                           int* __restrict__ lens) {
  __shared__ int red[256];
  const int blk = blockIdx.x;          // [0,32): t = blk>>4, b = blk&15
  const int t   = blk >> 4;
  const int b   = blk & 15;
  const int* m  = (t == 0 ? mask_x : mask_y) + (size_t)b * SEQ;
  const int tid = threadIdx.x;
  int s = 0;
  for (int i = tid; i < SEQ; i += 256) s += m[i];
  red[tid] = s;
  __syncthreads();
  for (int st = 128; st > 0; st >>= 1) {
    if (tid < st) red[tid] += red[tid + st];
    __syncthreads();
  }
  if (tid == 0) lens[blk] = SEQ - red[0];
}

// ---------------------------------------------------------------------------
// Kernel 2: streaming masked partial max (the HBM-bound part, ~128 MB read).
// 1024 blocks x 128 threads; each block streams one 64-row x 512-dim chunk
// with float4 (global_load_b128) fully-coalesced loads.  Prefix mask becomes
// a clamped trip count, so no divergence on the hot path.
// ---------------------------------------------------------------------------
__global__ void pmax_kernel(const float* __restrict__ x,
                            const float* __restrict__ y,
                            const int* __restrict__ lens,
                            float* __restrict__ partials) {
  const int blk = blockIdx.x;              // t*NB*NCHUNK + b*NCHUNK + c
  const int c   = blk % NCHUNK;
  const int tb  = blk / NCHUNK;            // t*16 + b
  const int b   = tb & 15;
  const int t   = tb >> 4;
  const float* __restrict__ src = (t == 0 ? x : y);

  int cnt = lens[tb] - c * CROWS;          // valid rows in this chunk
  if (cnt > CROWS) cnt = CROWS;

  const int d = threadIdx.x * 4;           // 128 threads * 4 dims = 512
  const float4* __restrict__ row =
      (const float4*)(src + ((size_t)b * SEQ + (size_t)c * CROWS) * DD + d);

  const float ninf = -__builtin_inff();
  float m0 = ninf, m1 = ninf, m2 = ninf, m3 = ninf;
#pragma unroll 8
  for (int s = 0; s < cnt; ++s) {
    float4 v = row[(size_t)s * (DD / 4)];
    m0 = fmaxf(m0, v.x);
    m1 = fmaxf(m1, v.y);
    m2 = fmaxf(m2, v.z);
    m3 = fmaxf(m3, v.w);
  }
  float4 out = make_float4(m0, m1, m2, m3);
  *(float4*)(partials + ((size_t)tb * NCHUNK + c) * DD + d) = out;
}

// ---------------------------------------------------------------------------
// Kernel 3: fold the 32 chunk-partials into inp[b][0:1024] = [x_in | y_in].
// 16 blocks x 256 threads; 2 MB of L2-resident traffic.
// ---------------------------------------------------------------------------
__global__ void fold_kernel(const float* __restrict__ partials,
                            float* __restrict__ inp) {
  const int b   = blockIdx.x;              // 16
  const int tid = threadIdx.x;             // 256
  const int t   = tid >> 7;                // 0: x half, 1: y half
  const int d   = (tid & 127) * 4;
  const int tb  = t * NB + b;
  const float4* __restrict__ p =
      (const float4*)(partials + (size_t)tb * NCHUNK * DD + d);
  const float ninf = -__builtin_inff();
  float4 m = make_float4(ninf, ninf, ninf, ninf);
#pragma unroll
  for (int c = 0; c < NCHUNK; ++c) {
    float4 v = p[(size_t)c * (DD / 4)];
    m.x = fmaxf(m.x, v.x);
    m.y = fmaxf(m.y, v.y);
    m.z = fmaxf(m.z, v.z);
    m.w = fmaxf(m.w, v.w);
  }
  *(float4*)(inp + (size_t)b * 1024 + t * DD + d) = m;
}

// ---------------------------------------------------------------------------
// Kernel 4: out = inp(16x1024) @ W^T + b via V_WMMA_F32_16X16X4_F32.
// One wave32.  A: M=batch rows, K-stepped by 4 (lanes 0-15 hold K=0,1;
// lanes 16-31 hold K=2,3 per the f32 16x4 A layout).  B: W in column N=0
// only (lane 0 -> B[0,0],B[1,0]; lane 16 -> B[2,0],B[3,0]); other lanes are
// zeroed via a hoisted 0/1 lane mask (branchless -> no EXEC juggling between
// WMMAs).  D column 0 then sits in lanes 0 and 16, VGPRs 0..7 = M 0..7/8..15.
// ---------------------------------------------------------------------------
__global__ void matvec_wmma_kernel(const float* __restrict__ inp,
                                   const float* __restrict__ W,
                                   const float* __restrict__ bias,
                                   float* __restrict__ out) {
  const int lane  = threadIdx.x;      // exactly 32 threads -> EXEC all ones
  const int row   = lane & 15;        // M index (batch)
  const int khalf = lane >> 4;        // 0 -> K pair {0,1}, 1 -> K pair {2,3}
  const float sel = (row == 0) ? 1.0f : 0.0f;   // one v_cndmask, hoisted

  const float* __restrict__ inpRow = inp + (size_t)row * 1024;

  v8f acc = {};
  for (int k0 = 0; k0 < 2 * DD; k0 += 4) {
    const int ka = k0 + khalf * 2;
    v2f a = *(const v2f*)(inpRow + ka);          // global_load_b64
    v2f w = *(const v2f*)(W + ka);               // uniform-per-half b64 load
    v2f bb;
    bb.x = w.x * sel;                            // column 0 of B only
    bb.y = w.y * sel;
    acc = __builtin_amdgcn_wmma_f32_16x16x4_f32(
        /*neg_a=*/false, a, /*neg_b=*/false, bb,
        /*c_mod=*/(short)0, acc, /*reuse_a=*/false, /*reuse_b=*/false);
  }
  if (row == 0) {                     // lane 0: M=0..7, lane 16: M=8..15
    const float bv = bias[0];
#pragma unroll
    for (int m = 0; m < 8; ++m) out[khalf * 8 + m] = acc[m] + bv;
  }
}

// ---------------------------------------------------------------------------
extern "C" void kernel_launch(void* const* d_in, const int* in_sizes, int n_in,
                              void* d_out, int out_size, void* d_ws, size_t ws_size,
                              hipStream_t stream) {
  (void)in_sizes; (void)n_in; (void)out_size; (void)ws_size;

  const float* x      = (const float*)d_in[0];   // (16, 2048, 512)
  const float* y      = (const float*)d_in[1];   // (16, 2048, 512)
  const int*   mask_x = (const int*)d_in[2];     // (16, 2048)
  const int*   mask_y = (const int*)d_in[3];     // (16, 2048)
  const float* W      = (const float*)d_in[4];   // (1, 1024)
  const float* bias   = (const float*)d_in[5];   // (1,)
  float*       out    = (float*)d_out;           // (16, 1)

  // Workspace layout (~2.16 MB):
  //   [0, 128)                      : lens (32 ints)
  //   [256, 256 + 2 MB)             : partials (2*16*32*512 f32)
  //   [.., .. + 64 KB)              : inp (16 x 1024 f32)
  char* ws = (char*)d_ws;
  int*   lens     = (int*)ws;
  float* partials = (float*)(ws + 256);
  float* inp      = (float*)(ws + 256 + (size_t)2 * NB * NCHUNK * DD * sizeof(float));

  len_kernel<<<2 * NB, 256, 0, stream>>>(mask_x, mask_y, lens);
  pmax_kernel<<<2 * NB * NCHUNK, 128, 0, stream>>>(x, y, lens, partials);
  fold_kernel<<<NB, 256, 0, stream>>>(partials, inp);
  matvec_wmma_kernel<<<1, 32, 0, stream>>>(inp, W, bias, out);
}